// DifferentiableModalPlate_76562087018697
// MI455X (gfx1250) — compile-verified
//
#include <hip/hip_runtime.h>
#include <math.h>

// ---------------------------------------------------------------------------
// Modal plate IR, factored as a GEMM:
//   disp[b*DT+dt] = sum_m A1[b,m]*C[m,dt] + A2[b,m]*S[m,dt]
// computed with V_WMMA_F32_16X16X4_F32 (fp32 WMMA: keeps the huge dynamic
// range of the decaying envelopes, which f16 WMMA would flush to zero).
// W is stored packed by K-pairs (float2) so each B fragment is one b64 load.
// ---------------------------------------------------------------------------

typedef float v2f __attribute__((ext_vector_type(2)));
typedef float v8f __attribute__((ext_vector_type(8)));

#define NMODES   6400            // 80x80 modes
#define KDIM     12800           // 2*NMODES (sin part | cos part)
#define KPAIRS   (KDIM/2)        // 6400 K-pairs
#define DT       128             // samples per time-block (GEMM N)
#define LXc      0.5f
#define KSTEP    (1.0f/44100.0f)
#define PI_F     3.14159265358979323846f

__device__ __forceinline__ float softplusf(float x) {
    return (x > 20.0f) ? x : log1pf(expf(x));
}
__device__ __forceinline__ float sigmoidf_(float x) {
    return 1.0f / (1.0f + expf(-x));
}

// ---------------------------------------------------------------------------
// Stage 1: per-mode omega, sigma, P' = P / (sin(omega*K)+1e-8)
// ---------------------------------------------------------------------------
__global__ void setup_modes(const float* mu_raw, const float* Dmu_raw,
                            const float* T0mu_raw, const float* Ly_raw,
                            const float* xo_raw, const float* yo_raw,
                            float* omega_o, float* sigma_o, float* Pp_o,
                            unsigned int* peak_o) {
    int mode = blockIdx.x * blockDim.x + threadIdx.x;
    if (mode == 0) *peak_o = 0u;              // reset peak accumulator per call
    if (mode >= NMODES) return;

    const float MAX_OM = 10000.0f * 2.0f * PI_F;
    const float MIN_OM = 20.0f * 2.0f * PI_F;
    const float ALPHA  = 1.15129254649702284e+00f;   // 3*ln10/tau0
    const float BETA   = 5.83251759816983398e-07f;   // 3*ln10*(1/tau1-1/tau0)/Om2sq

    float mu    = (softplusf(*mu_raw)   + 1e-4f) * 2.43f;
    float Dmu   = (softplusf(*Dmu_raw)  + 1e-4f) * 0.002452f;
    float T0mu  = (softplusf(*T0mu_raw) + 1e-4f) * 0.004115f;
    float Ly    = 1.1f + (4.0f - 1.1f) * sigmoidf_(*Ly_raw);
    float xo    = 0.49f * LXc + 0.51f * LXc * sigmoidf_(*xo_raw);
    float yo    = 0.51f * Ly  + 0.49f * Ly  * sigmoidf_(*yo_raw);
    float xi    = 0.1f * LXc;
    float yi    = 0.1f * Ly;

    float m = (float)(mode / 80 + 1);   // meshgrid 'ij': m varies slowly
    float n = (float)(mode % 80 + 1);

    float am = m * PI_F / LXc;
    float an = n * PI_F / Ly;
    float g1 = am * am + an * an;
    float omega_sq = T0mu * g1 + Dmu * g1 * g1;
    float omega = sqrtf(fmaxf(omega_sq, 0.0f));
    float valid = (omega <= MAX_OM && omega >= MIN_OM) ? 1.0f : 0.0f;

    float InW  = cosf(xi * PI_F * m / LXc) * cosf(yi * PI_F * n / Ly);
    float OutW = cosf(xo * PI_F * m / LXc) * cosf(yo * PI_F * n / Ly);
    float sigma = ALPHA + BETA * omega * omega;
    float ms    = 0.25f * mu * LXc * Ly;
    float P = OutW * InW * (KSTEP * KSTEP) * expf(-sigma * KSTEP) / ms * valid;
    float Pp = P / (sinf(omega * KSTEP) + 1e-8f);

    omega_o[mode] = omega;
    sigma_o[mode] = sigma;
    Pp_o[mode]    = Pp;
}

// ---------------------------------------------------------------------------
// Stage 2: packed W.  Logical W[KDIM x DT]:
//   rows 0..6399   : C[m,dt] = e^{-sigma*dt*K} * cos(dt*omega*K)
//   rows 6400..    : S[m,dt] = e^{-sigma*dt*K} * sin(dt*omega*K)
// Stored as Wp[p][dt] = (W[2p][dt], W[2p+1][dt]) : float2, p = 0..KPAIRS-1.
// One thread per (pair, dt): one coalesced 8-byte store.
// ---------------------------------------------------------------------------
__global__ void build_W(const float* __restrict__ omega,
                        const float* __restrict__ sigma,
                        v2f* __restrict__ Wp) {
    int idx = blockIdx.x * blockDim.x + threadIdx.x;
    if (idx >= KPAIRS * DT) return;
    int p  = idx / DT;
    int dt = idx - p * DT;
    int k0 = 2 * p;                         // k0, k0+1 lie in the same region
    int mm = (k0 < NMODES) ? k0 : k0 - NMODES;
    bool is_cos = (k0 < NMODES);
    float td = (float)dt;

    float w0 = omega[mm],     s0 = sigma[mm];
    float w1 = omega[mm + 1], s1 = sigma[mm + 1];
    float e0 = expf(-s0 * td * KSTEP);
    float e1 = expf(-s1 * td * KSTEP);
    float p0 = td * w0 * KSTEP;
    float p1 = td * w1 * KSTEP;

    v2f out;
    out.x = e0 * (is_cos ? cosf(p0) : sinf(p0));
    out.y = e1 * (is_cos ? cosf(p1) : sinf(p1));
    Wp[(size_t)p * DT + dt] = out;
}

// ---------------------------------------------------------------------------
// Stage 3: A[rowsA x KDIM] (GEMM A operand, row-major, stride KDIM)
//   cols 0..6399  : A1[b,m] = P'_m * e^{-sigma*(b*DT-1)*K} * sin(b*DT*omega*K)
//   cols 6400..   : A2[b,m] = P'_m * e^{-sigma*(b*DT-1)*K} * cos(b*DT*omega*K)
// ---------------------------------------------------------------------------
__global__ void build_A(const float* __restrict__ omega,
                        const float* __restrict__ sigma,
                        const float* __restrict__ Pp,
                        float* __restrict__ A, int rowsA) {
    int idx = blockIdx.x * blockDim.x + threadIdx.x;
    if (idx >= rowsA * KDIM) return;
    int row = idx / KDIM;
    int k   = idx - row * KDIM;
    int mm  = (k < NMODES) ? k : k - NMODES;
    float t0  = (float)(row * DT);
    float ph  = t0 * omega[mm] * KSTEP;
    float env = expf(-sigma[mm] * (t0 - 1.0f) * KSTEP);
    float tr  = (k < NMODES) ? sinf(ph) : cosf(ph);
    A[idx] = Pp[mm] * env * tr;
}

// ---------------------------------------------------------------------------
// Stage 4: GEMM  D[rowsA x DT] = A @ W  via V_WMMA_F32_16X16X4_F32.
// One 16x16 output tile per block; 8 waves split K=12800; LDS f32 atomics
// reduce the 8 partial accumulators.
// A fragment (16x4 fp32): lane%16 = M row, element K = 2*(lane/16) + vgpr
//   -> one b64 load from row-major A, pointer bump +16B per K-step.
// B fragment (4x16 fp32): lane%16 = N col, row  K = 2*(lane/16) + vgpr
//   -> one b64 load from K-pair-packed Wp, pointer bump +2048B per K-step.
// Explicit pointer induction so both sides fold into base+imm-offset clauses.
// ---------------------------------------------------------------------------
__global__ void __launch_bounds__(256)
gemm_wmma_f32(const float* __restrict__ A, const v2f* __restrict__ Wp,
              float* __restrict__ disp, int T) {
    __shared__ float tile[16 * 16];
    const int lane = threadIdx.x & 31;
    const int wave = threadIdx.x >> 5;       // 0..7
    const int lo   = lane & 15;              // M row (A) / N col (B)
    const int hi   = lane >> 4;              // 0/1 -> K pair select
    const int mrow0 = blockIdx.x * 16;       // time-block tile base
    const int ncol0 = blockIdx.y * 16;       // dt tile base

    tile[threadIdx.x] = 0.0f;
    __syncthreads();

    const int pchunk = KPAIRS / 8;           // 800 K-pairs per wave
    const int p0 = wave * pchunk;
    const int nsteps = pchunk / 2;           // 400 WMMA K-steps per wave

    // Per-lane streaming pointers (float2 units), constant stride per step.
    const v2f* ap = (const v2f*)(A + (size_t)(mrow0 + lo) * KDIM) + (p0 + hi);
    const v2f* bp = Wp + (size_t)(p0 + hi) * DT + (ncol0 + lo);

    v8f acc = {};
    #pragma unroll 8
    for (int i = 0; i < nsteps; ++i) {
        v2f a = *ap;
        v2f b = *bp;
        acc = __builtin_amdgcn_wmma_f32_16x16x4_f32(
            /*neg_a=*/false, a, /*neg_b=*/false, b,
            /*c_mod=*/(short)0, acc, /*reuse_a=*/false, /*reuse_b=*/false);
        ap += 2;            // +16 bytes
        bp += 2 * DT;       // +2048 bytes
    }

    // D layout: VGPR v, lane -> row = v + 8*(lane/16), col = lane%16
    #pragma unroll
    for (int v = 0; v < 8; ++v) {
        atomicAdd(&tile[(v + 8 * hi) * 16 + lo], acc[v]);   // ds_add_f32
    }
    __syncthreads();

    int r = threadIdx.x >> 4;
    int c = threadIdx.x & 15;
    int t = (mrow0 + r) * DT + (ncol0 + c);
    if (t < T) disp[t] = tile[threadIdx.x];
}

// ---------------------------------------------------------------------------
// Stage 5/6: peak-normalize. abs values are non-negative, so the float bit
// pattern is monotonic and integer atomicMax is exact.
// ---------------------------------------------------------------------------
__global__ void peak_abs(const float* __restrict__ disp, unsigned int* peak, int T) {
    int t = blockIdx.x * blockDim.x + threadIdx.x;
    if (t < T) atomicMax(peak, __float_as_uint(fabsf(disp[t])));
}

__global__ void normalize_k(const float* __restrict__ disp,
                            const unsigned int* __restrict__ peak,
                            float* __restrict__ out, int T) {
    int t = blockIdx.x * blockDim.x + threadIdx.x;
    if (t < T) out[t] = disp[t] / (__uint_as_float(*peak) + 1e-8f);
}

// ---------------------------------------------------------------------------
extern "C" void kernel_launch(void* const* d_in, const int* in_sizes, int n_in,
                              void* d_out, int out_size, void* d_ws, size_t ws_size,
                              hipStream_t stream) {
    const float* mu_raw   = (const float*)d_in[0];
    const float* Dmu_raw  = (const float*)d_in[1];
    const float* T0mu_raw = (const float*)d_in[2];
    const float* Ly_raw   = (const float*)d_in[3];
    const float* xo_raw   = (const float*)d_in[4];
    const float* yo_raw   = (const float*)d_in[5];
    // d_in[6] = num_samples (device int); use host-side out_size instead.

    const int T      = out_size;                  // 22050
    const int nb     = (T + DT - 1) / DT;         // time blocks (173)
    const int mtiles = (nb + 15) / 16;            // M tiles (11)
    const int rowsA  = mtiles * 16;               // padded rows (176)

    // Workspace layout (floats). Total ~15.7 MB for T=22050.
    float* ws      = (float*)d_ws;
    float* omega_b = ws;                               // 6400
    float* sigma_b = ws + NMODES;                      // 6400
    float* Pp_b    = ws + 2 * NMODES;                  // 6400
    unsigned int* peak = (unsigned int*)(ws + 3 * NMODES);
    float* A_b     = ws + 3 * NMODES + 256;            // rowsA * KDIM
    v2f*   Wp_b    = (v2f*)(A_b + (size_t)rowsA * KDIM);   // KPAIRS * DT float2
    float* disp_b  = (float*)(Wp_b + (size_t)KPAIRS * DT); // rowsA * DT

    setup_modes<<<(NMODES + 255) / 256, 256, 0, stream>>>(
        mu_raw, Dmu_raw, T0mu_raw, Ly_raw, xo_raw, yo_raw,
        omega_b, sigma_b, Pp_b, peak);

    build_W<<<(KPAIRS * DT + 255) / 256, 256, 0, stream>>>(omega_b, sigma_b, Wp_b);

    build_A<<<(rowsA * KDIM + 255) / 256, 256, 0, stream>>>(
        omega_b, sigma_b, Pp_b, A_b, rowsA);

    dim3 gemm_grid(mtiles, DT / 16);                   // 11 x 8 tiles
    gemm_wmma_f32<<<gemm_grid, 256, 0, stream>>>(A_b, Wp_b, disp_b, T);

    peak_abs<<<(T + 255) / 256, 256, 0, stream>>>(disp_b, peak, T);
    normalize_k<<<(T + 255) / 256, 256, 0, stream>>>(disp_b, peak, (float*)d_out, T);
}